// reaction_diffusion_nature_plus_90872918049156
// MI455X (gfx1250) — compile-verified
//
#include <hip/hip_runtime.h>
#include <math.h>

#define N_NODES 207
#define NP      208      // padded to 13*16
#define E_EDGES 1722
#define T_RES   12
#define L_LEN   2048
#define B_BATCH 64
#define NCHUNK  (NP / 16)   // 13 K-chunks

typedef __attribute__((ext_vector_type(2))) float        v2f;
typedef __attribute__((ext_vector_type(8))) float        v8f;
typedef __attribute__((ext_vector_type(4))) unsigned int u32x4;
typedef __attribute__((ext_vector_type(4))) int          i32x4;
typedef __attribute__((ext_vector_type(8))) int          i32x8;

// ---------------------------------------------------------------------------
// Kernel 1: build per-batch operator matrices in workspace (padded NPxNP).
//   ws[0*B + b] : RW   = diag(colsum(Ar))  - Ar
//   ws[1*B + b] : RWa  = diag(colsum(Ara)) + Ara   (b==0: diag term == 0)
//   ws[2*B + b] : DWc  = DW + DWa
// ---------------------------------------------------------------------------
__global__ __launch_bounds__(256)
void build_mats(const int* __restrict__ ind,
                const int* __restrict__ efrom,
                const int* __restrict__ eto,
                const float* __restrict__ wr,
                const float* __restrict__ wd,
                const float* __restrict__ wra,
                const float* __restrict__ wda,
                float* __restrict__ ws)
{
    const int b   = blockIdx.x;
    const int tid = threadIdx.x;
    const long mstride = (long)NP * NP;
    float* RW  = ws + (long)b * mstride;
    float* RWa = ws + (long)(B_BATCH + b) * mstride;
    float* DWc = ws + (long)(2 * B_BATCH + b) * mstride;

    // zero all three matrices (covers the padding rows/cols too)
    float4* z0 = (float4*)RW;
    float4* z1 = (float4*)RWa;
    float4* z2 = (float4*)DWc;
    const int n4 = (NP * NP) / 4;
    float4 z = {0.f, 0.f, 0.f, 0.f};
    for (int i = tid; i < n4; i += 256) { z0[i] = z; z1[i] = z; z2[i] = z; }
    __syncthreads();

    const int ii = ind[b] / T_RES;
    const float* wrp  = wr  + (long)ii * E_EDGES;
    const float* wdp  = wd  + (long)ii * E_EDGES;
    const float* wrap = wra + (long)ii * E_EDGES;
    const float* wdap = wda + (long)ii * E_EDGES;

    for (int e = tid; e < E_EDGES; e += 256) {
        const int f = efrom[e];
        const int t = eto[e];
        const float vr  = wrp[e];
        const float vd  = wdp[e];
        const float vra = wrap[e];
        const float vda = wdap[e];
        // RW = diag(colsum Ar) - Ar ; Ar[f,t]+=vr, colsum lands on column t
        atomicAdd(&RW[f * NP + t], -vr);
        atomicAdd(&RW[t * NP + t],  vr);
        // RWa = diag(colsum Ara) + Ara ; batch 0 diag = colsum(RW[0]) == 0
        atomicAdd(&RWa[f * NP + t], vra);
        if (b != 0) atomicAdd(&RWa[t * NP + t], vra);
        // DWc = (diag(colsum Ad) - Ad) + (diag(colsum Ada) + Ada)
        atomicAdd(&DWc[t * NP + f], vda - vd);
        atomicAdd(&DWc[f * NP + f], vda + vd);
    }
}

// ---------------------------------------------------------------------------
// TDM 2-D tile load: global (strided) -> LDS (packed, optional 17-float row
// padding via pad_interval=16 DWORDs / pad_amount=1 DWORD).
// Descriptor layout per CDNA5 ISA ch.8 (D# groups 0/1); 2-D tensor so
// groups 2/3 are zero. data_size=4 bytes. Rows beyond tensor_dim1 read 0.
// ---------------------------------------------------------------------------
__device__ static inline void tdm_load_2d(const void* gsrc, unsigned lds_off,
                                          unsigned dim0, unsigned dim1,
                                          unsigned tile0, unsigned tile1,
                                          unsigned stride_elems, bool pad17)
{
    const unsigned long long ga = (unsigned long long)(uintptr_t)gsrc;
    u32x4 g0;
    g0[0] = 1u;                                              // count=1 (valid)
    g0[1] = lds_off;                                         // lds_addr bytes
    g0[2] = (unsigned)(ga & 0xffffffffull);                  // global_addr lo
    g0[3] = (unsigned)((ga >> 32) & 0x01ffffffull)           // global_addr hi
          | (2u << 30);                                      // type=2 (image)

    i32x8 g1;
    unsigned w0 = (2u << 16);                                // data_size = 4B
    if (pad17) w0 |= (1u << 20) | (3u << 22);                // pad 1 DW / 16 DW
    g1[0] = (int)w0;
    g1[1] = (int)((dim0 & 0xffffu) << 16);                   // tensor_dim0 lo
    g1[2] = (int)(((dim0 >> 16) & 0xffffu) | ((dim1 & 0xffffu) << 16));
    g1[3] = (int)(((dim1 >> 16) & 0xffffu) | (tile0 << 16)); // tile_dim0
    g1[4] = (int)(tile1 & 0xffffu);                          // tile_dim1 (dim2=0)
    g1[5] = (int)stride_elems;                               // dim0_stride lo
    g1[6] = 0;                                               // stride hi
    g1[7] = 0;

    const i32x4 gz4 = {0, 0, 0, 0};
    const i32x8 gz8 = {0, 0, 0, 0, 0, 0, 0, 0};
    __builtin_amdgcn_tensor_load_to_lds(g0, g1, gz4, gz4, gz8, 0);
}

__device__ static inline unsigned lds_off_of(const void* p)
{
    // flat shared pointer: low 32 bits == LDS byte offset (LDS aperture maps
    // addr[31:0] to the wave's LDS allocation)
    return (unsigned)(uintptr_t)p;
}

// ---------------------------------------------------------------------------
// Kernel 2: out = tanh(RW@x+br) + tanh(RWa@x+bra) + DWc@x + bd + bda + x
// 256 threads = 8 waves; one batch / 16-row tile / 128-col L group per block.
// All LDS staging via double-buffered tensor_load_to_lds (wave0: x tile,
// wave1: three A tiles), compute via V_WMMA_F32_16X16X4_F32 chains.
// ---------------------------------------------------------------------------
__global__ __launch_bounds__(256)
void rd_matmul(const float* __restrict__ inputs,  // (B, 2, N, L), channel 0
               const int*   __restrict__ ind,
               const float* __restrict__ br,      // (T, N)
               const float* __restrict__ bd,
               const float* __restrict__ bra,
               const float* __restrict__ bda,
               const float* __restrict__ ws,
               float* __restrict__ out)           // (B, N, L)
{
    __shared__ float lA0[2][16][17];   // TDM pads rows to 17 floats
    __shared__ float lA1[2][16][17];
    __shared__ float lA2[2][16][17];
    __shared__ float lX [2][16][128];

    const int cg   = blockIdx.x;   // 0..15  (128-wide L group)
    const int rt   = blockIdx.y;   // 0..12  (16-row tile)
    const int b    = blockIdx.z;   // 0..63
    const int tid  = threadIdx.x;
    const int lane = tid & 31;
    const int w    = tid >> 5;

    const long mstride = (long)NP * NP;
    const float* RW  = ws + (long)b * mstride;
    const float* RWa = ws + (long)(B_BATCH + b) * mstride;
    const float* DWc = ws + (long)(2 * B_BATCH + b) * mstride;
    const float* xb  = inputs + (long)b * 2 * N_NODES * L_LEN;  // channel 0
    const int l0   = cg * 128;
    const int rt16 = rt * 16;

    v8f acc0 = {};
    v8f acc1 = {};
    v8f acc2 = {};

    // prologue: stage chunk 0
    if (w == 0) {
        tdm_load_2d(xb + l0, lds_off_of(&lX[0][0][0]),
                    L_LEN, N_NODES, 128, 16, L_LEN, false);
    } else if (w == 1) {
        tdm_load_2d(RW  + (long)rt16 * NP, lds_off_of(&lA0[0][0][0]),
                    NP, 16, 16, 16, NP, true);
        tdm_load_2d(RWa + (long)rt16 * NP, lds_off_of(&lA1[0][0][0]),
                    NP, 16, 16, 16, NP, true);
        tdm_load_2d(DWc + (long)rt16 * NP, lds_off_of(&lA2[0][0][0]),
                    NP, 16, 16, 16, NP, true);
    }

    const int mrow  = lane & 15;            // M (A) / N (B) within tile
    const int khalf = (lane >> 4) * 2;      // lanes 16..31 carry K=2,3

    for (int i = 0; i < NCHUNK; ++i) {
        const int cb = i & 1;
        const int nb = (i + 1) & 1;
        if (i + 1 < NCHUNK) {
            const int k1 = (i + 1) * 16;
            if (w == 0) {
                // rows beyond N_NODES-k1 zero-filled by TDM OOB handling
                tdm_load_2d(xb + (long)k1 * L_LEN + l0,
                            lds_off_of(&lX[nb][0][0]),
                            L_LEN, (unsigned)(N_NODES - k1), 128, 16,
                            L_LEN, false);
                __builtin_amdgcn_s_wait_tensorcnt(1);   // chunk i complete
            } else if (w == 1) {
                tdm_load_2d(RW  + (long)rt16 * NP + k1,
                            lds_off_of(&lA0[nb][0][0]), NP, 16, 16, 16, NP, true);
                tdm_load_2d(RWa + (long)rt16 * NP + k1,
                            lds_off_of(&lA1[nb][0][0]), NP, 16, 16, 16, NP, true);
                tdm_load_2d(DWc + (long)rt16 * NP + k1,
                            lds_off_of(&lA2[nb][0][0]), NP, 16, 16, 16, NP, true);
                __builtin_amdgcn_s_wait_tensorcnt(3);   // chunk i's 3 complete
            }
        } else {
            __builtin_amdgcn_s_wait_tensorcnt(0);
        }
        __syncthreads();

        #pragma unroll
        for (int kk = 0; kk < 16; kk += 4) {
            v2f a0, a1, a2, bx;
            a0.x = lA0[cb][mrow][kk + khalf]; a0.y = lA0[cb][mrow][kk + khalf + 1];
            a1.x = lA1[cb][mrow][kk + khalf]; a1.y = lA1[cb][mrow][kk + khalf + 1];
            a2.x = lA2[cb][mrow][kk + khalf]; a2.y = lA2[cb][mrow][kk + khalf + 1];
            bx.x = lX[cb][kk + khalf][w * 16 + mrow];
            bx.y = lX[cb][kk + khalf + 1][w * 16 + mrow];
            acc0 = __builtin_amdgcn_wmma_f32_16x16x4_f32(false, a0, false, bx,
                                                         (short)0, acc0, false, false);
            acc1 = __builtin_amdgcn_wmma_f32_16x16x4_f32(false, a1, false, bx,
                                                         (short)0, acc1, false, false);
            acc2 = __builtin_amdgcn_wmma_f32_16x16x4_f32(false, a2, false, bx,
                                                         (short)0, acc2, false, false);
        }
        __syncthreads();
    }

    // Epilogue. 16x16 f32 D layout: VGPR r -> M = r (lanes 0-15) / r+8
    // (lanes 16-31); N = lane & 15.
    const int ii = ind[b] / T_RES;
    const int l  = l0 + w * 16 + (lane & 15);
    const int mo = (lane >> 4) << 3;
    #pragma unroll
    for (int rr = 0; rr < 8; ++rr) {
        const int m = rt16 + rr + mo;
        if (m < N_NODES) {
            const float xv = xb[(long)m * L_LEN + l];
            const float o = tanhf(acc0[rr] + br[ii * N_NODES + m])
                          + tanhf(acc1[rr] + bra[ii * N_NODES + m])
                          + acc2[rr] + bd[ii * N_NODES + m] + bda[ii * N_NODES + m]
                          + xv;
            out[(long)b * N_NODES * L_LEN + (long)m * L_LEN + l] = o;
        }
    }
}

extern "C" void kernel_launch(void* const* d_in, const int* in_sizes, int n_in,
                              void* d_out, int out_size, void* d_ws, size_t ws_size,
                              hipStream_t stream) {
    const float* inputs = (const float*)d_in[0];   // (64, 2, 207, 2048)
    const int*   ind    = (const int*)  d_in[1];   // (64,)
    const int*   efrom  = (const int*)  d_in[2];   // (1722,)
    const int*   eto    = (const int*)  d_in[3];   // (1722,)
    const float* wr     = (const float*)d_in[4];   // (24, 1722)
    const float* wd     = (const float*)d_in[5];
    const float* wra    = (const float*)d_in[6];
    const float* wda    = (const float*)d_in[7];
    const float* br     = (const float*)d_in[8];   // (24, 207)
    const float* bd     = (const float*)d_in[9];
    const float* bra    = (const float*)d_in[10];
    const float* bda    = (const float*)d_in[11];
    float* out = (float*)d_out;
    float* ws  = (float*)d_ws;   // needs 3*64*208*208*4 ~= 33 MB

    build_mats<<<B_BATCH, 256, 0, stream>>>(ind, efrom, eto, wr, wd, wra, wda, ws);

    dim3 grid(L_LEN / 128, NP / 16, B_BATCH);    // (16, 13, 64)
    rd_matmul<<<grid, 256, 0, stream>>>(inputs, ind, br, bd, bra, bda, ws, out);
}